// MultiHeadedSelfAttention_53042846105796
// MI455X (gfx1250) — compile-verified
//
#include <hip/hip_runtime.h>
#include <hip/hip_bf16.h>

typedef float v2f __attribute__((ext_vector_type(2)));
typedef float v8f __attribute__((ext_vector_type(8)));
typedef int   v4i_vs __attribute__((vector_size(16)));

#define AS1 __attribute__((address_space(1)))
#define AS3 __attribute__((address_space(3)))

#define DDIM 1024
#define SS   2048
#define NB   2
#define NH   16
#define DH   64

#if defined(__has_builtin)
#if __has_builtin(__builtin_amdgcn_global_load_async_to_lds_b128) && \
    __has_builtin(__builtin_amdgcn_s_wait_asynccnt)
#define USE_ASYNC_LDS 1
#endif
#endif
#ifndef USE_ASYNC_LDS
#define USE_ASYNC_LDS 0
#endif

// ---------------------------------------------------------------------------
// QKV projection: out(B,H,S,Dh) = heads(x @ W + bias)
// block = 256 threads (8 waves as 2 row-groups x 4 col-groups).
// Block tile: 128 rows x 256 cols. Each wave: 64x64 = 4Mx4N accumulators,
// so every W B-operand is reused by 4 WMMAs (0.5 global loads / wmma).
// X k-chunks (128x32) staged in LDS via async-to-LDS (ASYNCcnt) when present.
// ---------------------------------------------------------------------------
__global__ __launch_bounds__(256) void qkv_proj_kernel(
    const float* __restrict__ x, const float* __restrict__ W,
    const float* __restrict__ bias, float* __restrict__ out) {
  __shared__ float xs[128][36];  // stride 36: 16B-aligned rows, conflict-free

  const int tid  = threadIdx.x;
  const int wave = tid >> 5;
  const int lane = tid & 31;
  const int ln   = lane & 15;
  const int half = lane >> 4;
  const int kb   = half * 2;          // K offset within 4-wide chunk
  const int rw   = wave >> 2;         // row-group 0..1
  const int cw   = wave & 3;          // col-group 0..3
  const int row0 = blockIdx.y * 128;
  const int rowW = rw * 64;           // wave row base within block
  const int colB = blockIdx.x * 256 + cw * 64;

  v8f acc[4][4];
#pragma unroll
  for (int mi = 0; mi < 4; ++mi)
#pragma unroll
    for (int ni = 0; ni < 4; ++ni)
#pragma unroll
      for (int i = 0; i < 8; ++i) acc[mi][ni][i] = 0.0f;

  for (int kc = 0; kc < DDIM / 32; ++kc) {
    __syncthreads();  // WAR: previous chunk fully consumed
#if USE_ASYNC_LDS
    {
#pragma unroll
      for (int q2 = 0; q2 < 4; ++q2) {
        const int lin = tid * 4 + q2 * 1024;
        const int r = lin >> 5;
        const int c = lin & 31;
        const float* gp = x + (size_t)(row0 + r) * DDIM + kc * 32 + c;
        __builtin_amdgcn_global_load_async_to_lds_b128(
            (AS1 v4i_vs*)(uintptr_t)gp,
            (AS3 v4i_vs*)(uint32_t)(uintptr_t)&xs[r][c], 0, 0);
      }
      __builtin_amdgcn_s_wait_asynccnt(0);
    }
#else
    {
#pragma unroll
      for (int q2 = 0; q2 < 4; ++q2) {
        const int lin = tid * 4 + q2 * 1024;
        const int r = lin >> 5;
        const int c = lin & 31;
        const float4 v =
            *(const float4*)(x + (size_t)(row0 + r) * DDIM + kc * 32 + c);
        xs[r][c + 0] = v.x; xs[r][c + 1] = v.y;
        xs[r][c + 2] = v.z; xs[r][c + 3] = v.w;
      }
    }
#endif
    __syncthreads();

#pragma unroll
    for (int kk = 0; kk < 8; ++kk) {
      v2f a[4];
#pragma unroll
      for (int mi = 0; mi < 4; ++mi) {
        a[mi].x = xs[rowW + mi * 16 + ln][kk * 4 + kb];
        a[mi].y = xs[rowW + mi * 16 + ln][kk * 4 + kb + 1];
      }
      const int kg = kc * 32 + kk * 4 + kb;
      const float* wp = W + (size_t)kg * DDIM + colB + ln;
#pragma unroll
      for (int ni = 0; ni < 4; ++ni) {
        v2f b;
        b.x = wp[ni * 16];          // W[kg  ][col]
        b.y = wp[DDIM + ni * 16];   // W[kg+1][col]
#pragma unroll
        for (int mi = 0; mi < 4; ++mi) {
          acc[mi][ni] = __builtin_amdgcn_wmma_f32_16x16x4_f32(
              false, a[mi], false, b, (short)0, acc[mi][ni], false, false);
        }
      }
    }
  }

  // bias + store into (B,H,S,Dh)
#pragma unroll
  for (int ni = 0; ni < 4; ++ni) {
    const int c  = colB + ni * 16 + ln;
    const float bv = bias[c];
    const int h  = c >> 6;
    const int dh = c & 63;
#pragma unroll
    for (int mi = 0; mi < 4; ++mi) {
#pragma unroll
      for (int i = 0; i < 8; ++i) {
        const int r  = row0 + rowW + mi * 16 + i + 8 * half;
        const int bb = r >> 11;   // batch
        const int s  = r & 2047;  // seq
        out[(((size_t)bb * NH + h) * SS + s) * DH + dh] = acc[mi][ni][i] + bv;
      }
    }
  }
}

// ---------------------------------------------------------------------------
// Flash attention: one wave per (32-query tile, b*H+h).
// Two 16-row M-subtiles share every K/V B-operand (1 vmem load / wmma).
// scores = Q K^T / 8, mask, streaming softmax, O = P V.
// ---------------------------------------------------------------------------
__global__ __launch_bounds__(32) void attn_kernel(
    const float* __restrict__ qbuf, const float* __restrict__ kbuf,
    const float* __restrict__ vbuf, const int* __restrict__ mask,
    float* __restrict__ out) {
  __shared__ float pl[32][20];  // P-tile transpose bounce, conflict-free

  const int lane = threadIdx.x;
  const int ln   = lane & 15;
  const int half = lane >> 4;
  const int kb   = half * 2;
  const int bh   = blockIdx.y;  // b*NH + h
  const int b    = bh >> 4;
  const int h    = bh & 15;
  const int q0   = blockIdx.x * 32;

  const float* Q = qbuf + (size_t)bh * SS * DH;
  const float* K = kbuf + (size_t)bh * SS * DH;
  const float* V = vbuf + (size_t)bh * SS * DH;
  const int* mk  = mask + b * SS;

  // Preload both 16-query A tiles (A layout: lane m=ln, K=kk*4+kb+{0,1}).
  v2f qa[2][16];
#pragma unroll
  for (int g = 0; g < 2; ++g)
#pragma unroll
    for (int kk = 0; kk < 16; ++kk) {
      const float* p = Q + (size_t)(q0 + g * 16 + ln) * DH + kk * 4 + kb;
      qa[g][kk].x = p[0];
      qa[g][kk].y = p[1];
    }

  v8f acc[2][4];
  float mrow[2][8], lrow[2][8];
#pragma unroll
  for (int g = 0; g < 2; ++g) {
#pragma unroll
    for (int j = 0; j < 4; ++j)
#pragma unroll
      for (int i = 0; i < 8; ++i) acc[g][j][i] = 0.0f;
#pragma unroll
    for (int i = 0; i < 8; ++i) { mrow[g][i] = -3.0e38f; lrow[g][i] = 0.0f; }
  }

  for (int t = 0; t < SS / 16; ++t) {
    const int k0 = t * 16;

    if (t + 4 < SS / 16) {  // warm L2/WGP$ ahead of the stream
      __builtin_prefetch(K + (size_t)(k0 + 64 + ln) * DH, 0, 1);
      __builtin_prefetch(V + (size_t)(k0 + 64 + ln) * DH, 0, 1);
    }

    // ---- score tiles: sc[g] = Q_g(16x64) x K^T(64x16) ----
    v8f sc[2];
#pragma unroll
    for (int g = 0; g < 2; ++g)
#pragma unroll
      for (int i = 0; i < 8; ++i) sc[g][i] = 0.0f;
#pragma unroll
    for (int kk = 0; kk < 16; ++kk) {
      v2f bv;
      const float* p = K + (size_t)(k0 + ln) * DH + kk * 4 + kb;
      bv.x = p[0];
      bv.y = p[1];
      sc[0] = __builtin_amdgcn_wmma_f32_16x16x4_f32(
          false, qa[0][kk], false, bv, (short)0, sc[0], false, false);
      sc[1] = __builtin_amdgcn_wmma_f32_16x16x4_f32(
          false, qa[1][kk], false, bv, (short)0, sc[1], false, false);
    }

    // ---- scale + mask + streaming softmax (lane owns key col k0+ln) ----
    const float pen = 10000.0f * (1.0f - (float)mk[k0 + ln]);
    float p8[2][8];
#pragma unroll
    for (int g = 0; g < 2; ++g) {
      float s8[8], mx[8];
#pragma unroll
      for (int i = 0; i < 8; ++i) {
        s8[i] = sc[g][i] * 0.125f - pen;  // 1/sqrt(64)
        mx[i] = s8[i];
      }
      for (int off = 1; off < 16; off <<= 1)
#pragma unroll
        for (int i = 0; i < 8; ++i)
          mx[i] = fmaxf(mx[i], __shfl_xor(mx[i], off, 32));
      float sum8[8], corr[8];
#pragma unroll
      for (int i = 0; i < 8; ++i) {
        const float mn = fmaxf(mrow[g][i], mx[i]);
        corr[i]    = __expf(mrow[g][i] - mn);
        mrow[g][i] = mn;
        p8[g][i]   = __expf(s8[i] - mn);
        sum8[i]    = p8[g][i];
      }
      for (int off = 1; off < 16; off <<= 1)
#pragma unroll
        for (int i = 0; i < 8; ++i)
          sum8[i] += __shfl_xor(sum8[i], off, 32);
#pragma unroll
      for (int i = 0; i < 8; ++i) lrow[g][i] = lrow[g][i] * corr[i] + sum8[i];
#pragma unroll
      for (int j = 0; j < 4; ++j)
#pragma unroll
        for (int i = 0; i < 8; ++i) acc[g][j][i] *= corr[i];
    }

    // ---- transpose P tiles through LDS into A-operand layout ----
    __syncthreads();  // WAR vs. previous tile's reads
#pragma unroll
    for (int g = 0; g < 2; ++g)
#pragma unroll
      for (int i = 0; i < 8; ++i) pl[g * 16 + i + 8 * half][ln] = p8[g][i];
    __syncthreads();

    // ---- O_g += P_g(16x16) x V(16x64), V operand shared by both g ----
#pragma unroll
    for (int kk = 0; kk < 4; ++kk) {
      v2f a0, a1;
      a0.x = pl[ln][kk * 4 + kb];
      a0.y = pl[ln][kk * 4 + kb + 1];
      a1.x = pl[16 + ln][kk * 4 + kb];
      a1.y = pl[16 + ln][kk * 4 + kb + 1];
      const float* vp = V + (size_t)(k0 + kk * 4 + kb) * DH + ln;
#pragma unroll
      for (int j = 0; j < 4; ++j) {
        v2f bv;
        bv.x = vp[j * 16];
        bv.y = vp[DH + j * 16];
        acc[0][j] = __builtin_amdgcn_wmma_f32_16x16x4_f32(
            false, a0, false, bv, (short)0, acc[0][j], false, false);
        acc[1][j] = __builtin_amdgcn_wmma_f32_16x16x4_f32(
            false, a1, false, bv, (short)0, acc[1][j], false, false);
      }
    }
  }

  // ---- normalize + store (B,S,D) ----
#pragma unroll
  for (int g = 0; g < 2; ++g)
#pragma unroll
    for (int j = 0; j < 4; ++j)
#pragma unroll
      for (int i = 0; i < 8; ++i) {
        const int r = q0 + g * 16 + i + 8 * half;
        out[((size_t)b * SS + r) * DDIM + h * 64 + j * 16 + ln] =
            acc[g][j][i] / lrow[g][i];
      }
}

// ---------------------------------------------------------------------------
extern "C" void kernel_launch(void* const* d_in, const int* in_sizes, int n_in,
                              void* d_out, int out_size, void* d_ws,
                              size_t ws_size, hipStream_t stream) {
  const float* x    = (const float*)d_in[0];
  const int*   mask = (const int*)d_in[1];
  const float* Wq   = (const float*)d_in[2];
  const float* bq   = (const float*)d_in[3];
  const float* Wk   = (const float*)d_in[4];
  const float* bk   = (const float*)d_in[5];
  const float* Wv   = (const float*)d_in[6];
  const float* bv   = (const float*)d_in[7];
  float* out = (float*)d_out;

  const size_t elems = (size_t)NB * SS * DDIM;  // 4,194,304 per tensor
  float* qb  = (float*)d_ws;
  float* kb2 = qb + elems;
  float* vb  = kb2 + elems;

  dim3 gProj(DDIM / 256, (NB * SS) / 128, 1);  // 4 x 32 blocks
  qkv_proj_kernel<<<gProj, 256, 0, stream>>>(x, Wq, bq, qb);
  qkv_proj_kernel<<<gProj, 256, 0, stream>>>(x, Wk, bk, kb2);
  qkv_proj_kernel<<<gProj, 256, 0, stream>>>(x, Wv, bv, vb);

  dim3 gAttn(SS / 32, NB * NH, 1);  // 64 q-tiles x 32 (b,h)
  attn_kernel<<<gAttn, 32, 0, stream>>>(qb, kb2, vb, mask, out);
}